// XTLoss_28243704938958
// MI455X (gfx1250) — compile-verified
//
#include <hip/hip_runtime.h>

// ---------------------------------------------------------------------------
// XTLoss for MI455X (gfx1250): fused warp + LCN + loss, one pass over HBM.
// 9x9 separable box filter implemented as banded-ones matrix products on
// V_WMMA_F32_16X16X4_F32 (wave32 WMMA, full fp32 accumulate).
// ---------------------------------------------------------------------------

typedef float v2f __attribute__((ext_vector_type(2)));
typedef float v8f __attribute__((ext_vector_type(8)));

#define TILE  32          // inner tile (pixels)
#define HALO  4           // 9x9 box => 4-pixel halo
#define REG   40          // TILE + 2*HALO
#define RSTR  49          // region/vertical row stride (odd vs 64 banks)
#define HSTR  33          // inner-tile row stride
#define N_    8
#define C_    3
#define H_    720
#define W_    1280
#define KKF   81.0f
#define EPSL  1e-6f

__global__ __launch_bounds__(256) void xtloss_tile_kernel(
    const float* __restrict__ left,
    const float* __restrict__ right,
    const float* __restrict__ disp,
    float* __restrict__ partials)
{
    // Fields: 0 = S_left, 1 = Q_left, 2 = S_recon, 3 = Q_recon
    __shared__ float sS[4][REG * RSTR];     // 40x49 region (S/Q, halo incl.)
    __shared__ float sV[4][TILE * RSTR];    // 32x49 vertical 9-tap sums
    __shared__ float sH[4][TILE * HSTR];    // 32x33 full 9x9 box sums
    __shared__ float sRec[C_][TILE * HSTR]; // warped recon, inner tile
    __shared__ float sLft[C_][TILE * HSTR]; // left, inner tile
    __shared__ float red[256];

    const int tid  = threadIdx.x;
    const int lane = tid & 31;
    const int wave = tid >> 5;
    const int n    = blockIdx.z;
    const int h0   = blockIdx.y * TILE;
    const int w0   = blockIdx.x * TILE;

    // ---- Phase A: bilinear warp + per-pixel channel sums on 40x40 region ----
    for (int idx = tid; idx < REG * REG; idx += 256) {
        const int ry = idx / REG, rx = idx - ry * REG;
        const int h = h0 + ry - HALO;
        const int w = w0 + rx - HALO;
        float Sl = 0.f, Ql = 0.f, Sr = 0.f, Qr = 0.f;
        float recC[C_] = {0.f, 0.f, 0.f};
        float lftC[C_] = {0.f, 0.f, 0.f};
        if (h >= 0 && h < H_ && w >= 0 && w < W_) {
            const size_t pix = (size_t)(n * H_ + h) * W_ + w;
            const float d  = disp[pix];
            const float ix = (float)w - d;
            const float x0 = floorf(ix);
            const float wx = ix - x0;
            const int x0i = (int)x0;
            const int x1i = x0i + 1;
            const float v0 = (x0i >= 0 && x0i < W_) ? 1.0f : 0.0f;
            const float v1 = (x1i >= 0 && x1i < W_) ? 1.0f : 0.0f;
            const int i0 = min(max(x0i, 0), W_ - 1);
            const int i1 = min(max(x1i, 0), W_ - 1);
            const float wA = v0 * (1.0f - wx), wB = v1 * wx;
            #pragma unroll
            for (int c = 0; c < C_; ++c) {
                const size_t base = ((size_t)(n * C_ + c) * H_ + h) * W_;
                const float g0 = right[base + i0];
                const float g1 = right[base + i1];
                const float r  = g0 * wA + g1 * wB;   // warped recon
                const float l  = left[base + w];
                recC[c] = r; lftC[c] = l;
                Sl += l; Ql += l * l;
                Sr += r; Qr += r * r;
            }
        }
        sS[0][ry * RSTR + rx] = Sl;
        sS[1][ry * RSTR + rx] = Ql;
        sS[2][ry * RSTR + rx] = Sr;
        sS[3][ry * RSTR + rx] = Qr;
        if (rx >= HALO && rx < HALO + TILE && ry >= HALO && ry < HALO + TILE) {
            const int iy = ry - HALO, ixp = rx - HALO;
            #pragma unroll
            for (int c = 0; c < C_; ++c) {
                sRec[c][iy * HSTR + ixp] = recC[c];
                sLft[c][iy * HSTR + ixp] = lftC[c];
            }
        }
    }
    // Zero padding columns 40..47 (touched by the third vertical col-tile).
    for (int idx = tid; idx < REG * 8; idx += 256) {
        const int ry = idx >> 3, rx = REG + (idx & 7);
        sS[0][ry * RSTR + rx] = 0.f; sS[1][ry * RSTR + rx] = 0.f;
        sS[2][ry * RSTR + rx] = 0.f; sS[3][ry * RSTR + rx] = 0.f;
    }
    __syncthreads();

    // WMMA fragment lane decomposition (ISA 7.12.2, 32-bit A 16x4 / D 16x16):
    //   A: lane -> M = lane&15 ; half-wave selects K pair {0,1} or {2,3}
    //   B: lane -> N = lane&15 ; half-wave selects K pair
    //   D: lane -> N = lane&15 ; VGPR v -> M = v + 8*(lane>=16)
    const int ln16 = lane & 15;
    const int kb   = (lane >> 4) << 1;  // 0 or 2
    const int hi   = lane >> 4;

    // ---- Phase B: vertical 9-tap sum via banded WMMA ----
    // Y[m][n] = sum_{c=m..m+8} region[r0+c][c0+n]  (band matrix A, data B)
    // 24 tasks = 4 fields x 2 row-tiles x 3 col-tiles; 3 per wave (uniform).
    #pragma unroll
    for (int t = 0; t < 3; ++t) {
        const int task = wave + t * 8;        // 0..23
        const int f  = task & 3;
        const int rc = task >> 2;             // 0..5
        const int r0 = (rc & 1) * 16;         // 0,16 (output & input row base)
        const int c0 = (rc >> 1) * 16;        // 0,16,32
        const float* __restrict__ src = sS[f];
        v8f acc = {};
        #pragma unroll
        for (int ch = 0; ch < 6; ++ch) {      // 24 input rows / K=4
            const int cr = ch * 4 + kb;
            v2f a, b;
            a.x = (cr     >= ln16 && cr     <= ln16 + 8) ? 1.0f : 0.0f;
            a.y = (cr + 1 >= ln16 && cr + 1 <= ln16 + 8) ? 1.0f : 0.0f;
            b.x = src[(r0 + cr    ) * RSTR + c0 + ln16];
            b.y = src[(r0 + cr + 1) * RSTR + c0 + ln16];
            acc = __builtin_amdgcn_wmma_f32_16x16x4_f32(
                false, a, false, b, (short)0, acc, false, false);
        }
        #pragma unroll
        for (int v = 0; v < 8; ++v)
            sV[f][(r0 + v + 8 * hi) * RSTR + c0 + ln16] = acc[v];
    }
    __syncthreads();

    // ---- Phase C: horizontal 9-tap sum via WMMA (data A, band matrix B) ----
    // Y[m][n] = sum_{c=n..n+8} vbuf[r0+m][j0+c]
    // 16 tasks = 4 fields x 2 row-tiles x 2 col-tiles; 2 per wave (uniform).
    #pragma unroll
    for (int t = 0; t < 2; ++t) {
        const int task = wave + t * 8;        // 0..15
        const int f  = task & 3;
        const int rc = task >> 2;             // 0..3
        const int r0 = (rc & 1) * 16;
        const int j0 = (rc >> 1) * 16;
        const float* __restrict__ src = sV[f];
        v8f acc = {};
        #pragma unroll
        for (int ch = 0; ch < 6; ++ch) {
            const int cr = ch * 4 + kb;
            v2f a, b;
            a.x = src[(r0 + ln16) * RSTR + j0 + cr];
            a.y = src[(r0 + ln16) * RSTR + j0 + cr + 1];
            b.x = (cr     >= ln16 && cr     <= ln16 + 8) ? 1.0f : 0.0f;
            b.y = (cr + 1 >= ln16 && cr + 1 <= ln16 + 8) ? 1.0f : 0.0f;
            acc = __builtin_amdgcn_wmma_f32_16x16x4_f32(
                false, a, false, b, (short)0, acc, false, false);
        }
        #pragma unroll
        for (int v = 0; v < 8; ++v)
            sH[f][(r0 + v + 8 * hi) * HSTR + j0 + ln16] = acc[v];
    }
    __syncthreads();

    // ---- Phase D: LCN + loss on the inner 32x32 tile, block reduction ----
    float accum = 0.0f;
    for (int idx = tid; idx < TILE * TILE; idx += 256) {
        const int iy = idx >> 5, ixp = idx & 31;
        const int h = h0 + iy;                // W_ divides evenly; only H masks
        if (h < H_) {
            const float mL = sH[0][iy * HSTR + ixp] * (1.0f / KKF);
            const float qL = sH[1][iy * HSTR + ixp] * (1.0f / KKF);
            const float mR = sH[2][iy * HSTR + ixp] * (1.0f / KKF);
            const float qR = sH[3][iy * HSTR + ixp] * (1.0f / KKF);
            const float stdL = (qL - mL * mL) * (KKF / (KKF - 1.0f));
            const float stdR = (qR - mR * mR) * (KKF / (KKF - 1.0f));
            const float invL = 1.0f / (stdL + EPSL);
            const float invR = 1.0f / (stdR + EPSL);
            #pragma unroll
            for (int c = 0; c < C_; ++c) {
                const float l = sLft[c][iy * HSTR + ixp];
                const float r = sRec[c][iy * HSTR + ixp];
                accum += fabsf(((l - mL) * invL - (r - mR) * invR) * stdL);
            }
        }
    }
    red[tid] = accum;
    __syncthreads();
    #pragma unroll
    for (int s = 128; s > 0; s >>= 1) {
        if (tid < s) red[tid] += red[tid + s];
        __syncthreads();
    }
    if (tid == 0) {
        partials[(blockIdx.z * gridDim.y + blockIdx.y) * gridDim.x + blockIdx.x]
            = red[0];
    }
}

__global__ __launch_bounds__(256) void xtloss_reduce_kernel(
    const float* __restrict__ partials, int nparts, float* __restrict__ out)
{
    __shared__ float red[256];
    float s = 0.0f;
    for (int i = threadIdx.x; i < nparts; i += 256) s += partials[i];
    red[threadIdx.x] = s;
    __syncthreads();
    #pragma unroll
    for (int k = 128; k > 0; k >>= 1) {
        if (threadIdx.x < k) red[threadIdx.x] += red[threadIdx.x + k];
        __syncthreads();
    }
    if (threadIdx.x == 0)
        out[0] = red[0] * (1.0f / ((float)N_ * C_ * H_ * W_));
}

extern "C" void kernel_launch(void* const* d_in, const int* in_sizes, int n_in,
                              void* d_out, int out_size, void* d_ws, size_t ws_size,
                              hipStream_t stream) {
    const float* left  = (const float*)d_in[0];   // [8,3,720,1280] f32
    const float* right = (const float*)d_in[1];   // [8,3,720,1280] f32
    const float* disp  = (const float*)d_in[2];   // [8,720,1280]  f32
    float* out      = (float*)d_out;              // scalar f32
    float* partials = (float*)d_ws;               // nblocks floats (< 30 KB)

    dim3 grid(W_ / TILE, (H_ + TILE - 1) / TILE, N_);  // 40 x 23 x 8 = 7360
    xtloss_tile_kernel<<<grid, 256, 0, stream>>>(left, right, disp, partials);
    const int nparts = grid.x * grid.y * grid.z;
    xtloss_reduce_kernel<<<1, 256, 0, stream>>>(partials, nparts, out);
}